// Neighbor_69947837383445
// MI455X (gfx1250) — compile-verified
//
#include <hip/hip_runtime.h>

typedef float v2f __attribute__((ext_vector_type(2)));
typedef float v8f __attribute__((ext_vector_type(8)));

#define E_ROWS 32768
#define DA_COLS 1024
#define LOG2E 1.44269504088896340736f

// g(s) = s * f(s) for the three nonlinearities.
// NID 0: sigmoid  f = 1/(1+exp(-s))
// NID 1: tanh     f = 2/(1+exp(-2s)) - 1
// NID 2: relu     f = max(s,0)  ->  g = s>0 ? s*s : 0
template <int NID>
__device__ __forceinline__ float g_term(float s) {
  if (NID == 0) {
    float e = __builtin_amdgcn_exp2f(-LOG2E * s);          // v_exp_f32
    return s * __builtin_amdgcn_rcpf(1.0f + e);            // v_rcp_f32
  } else if (NID == 1) {
    float e = __builtin_amdgcn_exp2f(-2.0f * LOG2E * s);
    float r = __builtin_amdgcn_rcpf(1.0f + e);
    return s * (2.0f * r - 1.0f);
  } else {
    return s > 0.0f ? s * s : 0.0f;
  }
}

template <int NID>
__device__ __forceinline__ float row_accumulate(const float* __restrict__ wr,
                                                const float* xs, int lane) {
  float acc = 0.0f;
#pragma unroll
  for (int i = 0; i < 8; ++i) {
    int c = i * 128 + lane * 4;                 // 512B contiguous per wave per iter
    float4 w4 = *(const float4*)(wr + c);       // global_load_b128
    float4 x4 = *(const float4*)(xs + c);       // ds_load_b128
    acc += g_term<NID>(x4.x * w4.x);
    acc += g_term<NID>(x4.y * w4.y);
    acc += g_term<NID>(x4.z * w4.z);
    acc += g_term<NID>(x4.w * w4.w);
  }
  return acc;
}

__global__ __launch_bounds__(256) void edge_scores_kernel(
    const float* __restrict__ x, const float* __restrict__ ampli,
    const float* __restrict__ W, const int* __restrict__ nonlin_id,
    float* __restrict__ scores) {
  __shared__ float xs[DA_COLS];
  for (int i = threadIdx.x; i < DA_COLS; i += 256) xs[i] = x[i];
  __syncthreads();

  const int lane = threadIdx.x & 31;
  const int wave = threadIdx.x >> 5;
  const int r = blockIdx.x * 8 + wave;            // one row per wave32
  const float* wr = W + (size_t)r * DA_COLS;

  // nonlin_id is uniform per row -> force to SGPR, take a scalar branch
  const int nid = __builtin_amdgcn_readfirstlane(nonlin_id[r]);

  float acc;
  if (nid == 0)      acc = row_accumulate<0>(wr, xs, lane);
  else if (nid == 1) acc = row_accumulate<1>(wr, xs, lane);
  else               acc = row_accumulate<2>(wr, xs, lane);

  // ---- Cross-lane reduction via V_WMMA_F32_16X16X4_F32 with B = ones ----
  // A layout (16x4 f32): vgpr0 = {K0 | K2}, vgpr1 = {K1 | K3} across lane halves.
  // A vgpr0 = acc, vgpr1 = 0  =>  A[m,0]=acc[m], A[m,2]=acc[m+16].
  // With B = ones: D[m,n] = acc[m] + acc[m+16] for every n.
  v2f A; A[0] = acc;  A[1] = 0.0f;
  v2f B; B[0] = 1.0f; B[1] = 1.0f;
  v8f C = {};
  v8f D = __builtin_amdgcn_wmma_f32_16x16x4_f32(
      /*neg_a=*/false, A, /*neg_b=*/false, B,
      /*c_mod=*/(short)0, C, /*reuse_a=*/false, /*reuse_b=*/false);
  // D vgpr v: lanes 0-15 hold M=v, lanes 16-31 hold M=v+8. Sum the 8 vgprs:
  float p = D[0] + D[1] + D[2] + D[3] + D[4] + D[5] + D[6] + D[7];
  // lanes<16 now hold sum over m=0..7, lanes>=16 over m=8..15 -> one swap
  float total = p + __shfl_xor(p, 16, 32);

  if (lane == 0) scores[r] = total * ampli[0];
}

__global__ __launch_bounds__(1024) void softmax_kernel(
    const float* __restrict__ scores, float* __restrict__ out) {
  __shared__ float red[1024];
  const int tid = threadIdx.x;

  // pass 1: max
  float m = -3.402823466e38f;
  for (int i = tid; i < E_ROWS; i += 1024) m = fmaxf(m, scores[i]);
  red[tid] = m;
  __syncthreads();
  for (int s = 512; s > 0; s >>= 1) {
    if (tid < s) red[tid] = fmaxf(red[tid], red[tid + s]);
    __syncthreads();
  }
  const float gmax = red[0];
  __syncthreads();

  // pass 2: exp + sum (exp2 with log2e scale == exp, consistently applied)
  float sum = 0.0f;
  for (int i = tid; i < E_ROWS; i += 1024) {
    float e = __builtin_amdgcn_exp2f(LOG2E * (scores[i] - gmax));
    out[i] = e;
    sum += e;
  }
  red[tid] = sum;
  __syncthreads();
  for (int s = 512; s > 0; s >>= 1) {
    if (tid < s) red[tid] += red[tid + s];
    __syncthreads();
  }
  const float inv = 1.0f / red[0];

  // pass 3: normalize
  for (int i = tid; i < E_ROWS; i += 1024) out[i] *= inv;
}

extern "C" void kernel_launch(void* const* d_in, const int* in_sizes, int n_in,
                              void* d_out, int out_size, void* d_ws, size_t ws_size,
                              hipStream_t stream) {
  const float* x      = (const float*)d_in[0];
  const float* ampli  = (const float*)d_in[1];
  const float* W      = (const float*)d_in[2];
  const int*   nid    = (const int*)d_in[3];
  float* scores = (float*)d_ws;                 // 32768 floats of scratch
  float* out    = (float*)d_out;

  edge_scores_kernel<<<E_ROWS / 8, 256, 0, stream>>>(x, ampli, W, nid, scores);
  softmax_kernel<<<1, 1024, 0, stream>>>(scores, out);
}